// ODEINDLayer_17609365913953
// MI455X (gfx1250) — compile-verified
//
#include <hip/hip_runtime.h>
#include <hip/hip_bf16.h>

// ---------------- problem constants (from reference) ----------------
constexpr int BSZ  = 256;
constexpr int NID  = 16;
constexpr int T    = 50;
constexpr int K    = 2;
constexpr int KP   = K + 1;          // 3
constexpr int Bn   = BSZ * NID;      // 4096 batches
constexpr int NU   = T * KP;         // 150
constexpr int Nn   = 1 + NU;         // 151 (cols of A)
constexpr int M_IV = 1;
constexpr int M_D  = (T - 1) * K;    // 98
constexpr int Mm   = M_IV + T + M_D; // 149 (rows of A / KKT dim)
constexpr int DSTART = M_IV + T;     // 51: first "d" row (has 1 in col 0)

constexpr float GAMMA  = 0.5f;
constexpr float ALPHA  = 0.1f;
constexpr float JITTER = 1e-6f;

// ---------------- LDS geometry ----------------
constexpr int P    = 156;            // row pitch in floats (156 mod 64 = 28 -> few bank conflicts)
constexpr int ROWS = 160;            // 10 tiles of 16
constexpr int NT   = 10;             // 16x16 tile grid per dim
constexpr int NTILES_LOWER = NT * (NT + 1) / 2;  // 55
constexpr int KPAD = 152;            // K-extent for WMMA loop (cols padded to mult of 4)

typedef float v2f __attribute__((ext_vector_type(2)));
typedef float v8f __attribute__((ext_vector_type(8)));

__global__ __launch_bounds__(256, 1)
void ode_kkt_solve_kernel(const float* __restrict__ coeffs,
                          const float* __restrict__ rhs,
                          const float* __restrict__ iv_rhs,
                          const float* __restrict__ steps,
                          float* __restrict__ out)
{
    extern __shared__ float smem[];
    float* sA  = smem;                 // [ROWS][P]  dense A (padded, zero elsewhere)
    float* sG  = smem + ROWS * P;      // [ROWS][P]  Gram / KKT / LDL^T factor (lower)
    float* sV  = sG  + ROWS * P;       // [ROWS]     b -> z -> w -> lambda
    float* sDi = sV  + ROWS;           // [ROWS]     1 / D[k]

    const int b    = blockIdx.x;       // one batch per workgroup
    const int tid  = threadIdx.x;
    const int lane = tid & 31;
    const int wave = tid >> 5;

    // ---------- 1) zero A and b ----------
    for (int i = tid; i < ROWS * P; i += 256) sA[i] = 0.0f;
    for (int i = tid; i < ROWS; i += 256)     sV[i] = 0.0f;
    __syncthreads();

    // ---------- 2) sparse fill of A and b ----------
    if (tid == 0) {
        sA[0 * P + 1] = 1.0f;              // IV row: e_1  (NIVS=NIV=1)
        sV[0] = iv_rhs[b];
    }
    if (tid < T * KP) {                     // eq rows: row (1+t) has c[t, j] at col 1 + t*KP + j
        int t = tid / KP, j = tid % KP;
        sA[(M_IV + t) * P + 1 + t * KP + j] =
            coeffs[((size_t)b * T + t) * KP + j];
    }
    if (tid < T) {                          // b for eq rows
        sV[M_IV + tid] = rhs[(size_t)b * T + tid];
    }
    if (tid < M_D) {                        // d rows: col0 = 1; D0 - 0.5*h*SH pattern
        int trow = tid / K, j = tid % K;
        int m = DSTART + tid;
        float h = steps[(size_t)b * (T - 1) + trow];
        float* row = &sA[m * P];
        row[0] = 1.0f;
        row[1 + trow * KP + j]           = -1.0f;        // u(t,   j)
        row[1 + (trow + 1) * KP + j]     =  1.0f;        // u(t+1, j)
        row[1 + trow * KP + (j + 1)]     = -0.5f * h;    // u(t,   j+1)
        row[1 + (trow + 1) * KP + (j+1)] = -0.5f * h;    // u(t+1, j+1)
    }
    __syncthreads();

    // ---------- 3) G = A * A^T  via V_WMMA_F32_16X16X4_F32 (lower tiles only) ----------
    // Fragment layouts per CDNA5 ISA 7.12.2 (f32, 16x16x4):
    //   A (16x4):  lane%16 = M row; VGPR{0,1} hold K = {0,1} (lanes 0-15) / {2,3} (lanes 16-31)
    //   B (4x16):  lane%16 = N col; VGPR{0,1} hold K = {0,1} (lanes 0-15) / {2,3} (lanes 16-31)
    //   D (16x16): lane%16 = N; VGPR r -> M = r (lanes 0-15) or 8+r (lanes 16-31)
    for (int tile = wave; tile < NTILES_LOWER; tile += 8) {   // wave-uniform control flow
        int ti = 0, rem = tile;
        while (rem >= ti + 1) { rem -= ti + 1; ti++; }
        const int tj = rem;                                   // tj <= ti

        const int rA   = (ti * 16 + (lane & 15)) * P;   // A-fragment row base
        const int rB   = (tj * 16 + (lane & 15)) * P;   // B-fragment = rows of A for tile tj
        const int ksel = (lane >> 4) * 2;

        v8f acc = {};
        for (int k0 = 0; k0 < KPAD; k0 += 4) {
            const int ka = k0 + ksel;
            v2f a, bm;
            a.x  = sA[rA + ka];  a.y  = sA[rA + ka + 1];
            bm.x = sA[rB + ka];  bm.y = sA[rB + ka + 1];
#if __has_builtin(__builtin_amdgcn_wmma_f32_16x16x4_f32)
            acc = __builtin_amdgcn_wmma_f32_16x16x4_f32(
                    false, a, false, bm, (short)0, acc, false, false);
#else
            // scalar fallback with identical accumulator layout
            {
                const int n = lane & 15;
                const int mb = (lane >> 4) ? 8 : 0;
                #pragma unroll
                for (int r = 0; r < 8; r++) {
                    const int mrow = (ti * 16 + mb + r) * P;
                    float s = acc[r];
                    #pragma unroll
                    for (int kk = 0; kk < 4; kk++)
                        s += sA[mrow + k0 + kk] * sA[(tj * 16 + n) * P + k0 + kk];
                    acc[r] = s;
                }
            }
#endif
        }
        const int n  = lane & 15;
        const int mb = ti * 16 + ((lane >> 4) ? 8 : 0);
        #pragma unroll
        for (int r = 0; r < 8; r++)
            sG[(mb + r) * P + tj * 16 + n] = acc[r];
    }
    __syncthreads();

    // ---------- 4) KKT = G/alpha + (1/gamma - 1/alpha) * 1_d 1_d^T + jitter*I (lower) ----------
    {
        const float invA = 1.0f / ALPHA;
        const float corr = 1.0f / GAMMA - 1.0f / ALPHA;   // from col-0 (eps) scaling
        for (int i = tid; i < Mm; i += 256) {
            for (int j = 0; j <= i; j++) {
                float g = sG[i * P + j] * invA;
                if (i >= DSTART && j >= DSTART) g += corr;
                if (i == j) g += JITTER;
                sG[i * P + j] = g;
            }
        }
    }
    __syncthreads();

    // ---------- 5) in-place LDL^T (one barrier per column; writes only strict trailing) ----------
    for (int k = 0; k < Mm - 1; k++) {
        __syncthreads();
        const float invd = 1.0f / sG[k * P + k];
        const int i = k + 1 + tid;               // one trailing row per thread
        if (i < Mm) {
            const float gik = sG[i * P + k] * invd;
            for (int j = k + 1; j <= i; j++)
                sG[i * P + j] -= gik * sG[j * P + k];
        }
    }
    __syncthreads();
    for (int i = tid; i < Mm; i += 256) sDi[i] = 1.0f / sG[i * P + i];

    // ---------- 6) forward solve L z = b (L unit-lower, L[i][j] = G[i][j]*Di[j]) ----------
    for (int j = 0; j < Mm - 1; j++) {
        __syncthreads();
        const float vj = sV[j] * sDi[j];
        const int i = j + 1 + tid;
        if (i < Mm) sV[i] -= sG[i * P + j] * vj;
    }
    __syncthreads();
    for (int i = tid; i < Mm; i += 256) sV[i] *= sDi[i];   // w = D^-1 z

    // ---------- 7) backward solve L^T lam = w ----------
    for (int j = Mm - 1; j >= 1; j--) {
        __syncthreads();
        const float vj = sV[j];                  // = lambda[j]
        if (tid < j) sV[tid] -= sG[j * P + tid] * sDi[tid] * vj;
    }
    __syncthreads();

    // ---------- 8) x = qinv .* (A^T lam); scatter outputs ----------
    const size_t sizeU = (size_t)Bn * T;         // 204800 per u-plane
    if (tid < Nn) {
        const int n = tid;
        float s = 0.0f;
        for (int m = 0; m < Mm; m++) s += sA[m * P + n] * sV[m];
        const float x = s * (n == 0 ? (1.0f / GAMMA) : (1.0f / ALPHA));
        if (n == 0) {
            out[3 * sizeU + b] = x;                          // eps
        } else {
            const int nu = n - 1, tu = nu / KP, ju = nu % KP;
            out[(size_t)ju * sizeU + (size_t)b * T + tu] = x; // u0|u1|u2
        }
    }
    // steps passthrough: out[..., b, :, 0] = steps
    for (int i = tid; i < T - 1; i += 256)
        out[3 * sizeU + Bn + (size_t)b * (T - 1) + i] =
            steps[(size_t)b * (T - 1) + i];
}

extern "C" void kernel_launch(void* const* d_in, const int* in_sizes, int n_in,
                              void* d_out, int out_size, void* d_ws, size_t ws_size,
                              hipStream_t stream) {
    (void)in_sizes; (void)n_in; (void)out_size; (void)d_ws; (void)ws_size;
    const float* coeffs = (const float*)d_in[0];
    const float* rhs    = (const float*)d_in[1];
    const float* iv_rhs = (const float*)d_in[2];
    const float* steps  = (const float*)d_in[3];
    float* out = (float*)d_out;

    constexpr size_t smem_bytes =
        (size_t)(ROWS * P) * 2 * sizeof(float) + (size_t)ROWS * 2 * sizeof(float); // ~196 KB < 320 KB/WGP

    ode_kkt_solve_kernel<<<Bn, 256, smem_bytes, stream>>>(coeffs, rhs, iv_rhs, steps, out);
}